// NeRFVoxel_75539884802270
// MI455X (gfx1250) — compile-verified
//
#include <hip/hip_runtime.h>
#include <hip/hip_bf16.h>

#define RESO  128
#define IMG   192
#define NSTEP 64
#define TILE  16
#define ROWF  (TILE * 6)            // floats per tile row (16 px * 6) = 96
#define IMG_ROWF (IMG * 6)          // floats per image row = 1152

typedef unsigned int u32x4 __attribute__((ext_vector_type(4)));
typedef int          i32x4 __attribute__((ext_vector_type(4)));
typedef int          i32x8 __attribute__((ext_vector_type(8)));

// Compile-time constants, computed in double then rounded once to f32,
// matching numpy/JAX semantics (python-float op against f32 array).
__device__ __forceinline__ float k_G()    { return 1.3f; }
__device__ __forceinline__ float k_VL()   { return (float)(2.0 * 1.3 / 128.0); }
__device__ __forceinline__ float k_HALF() { return (float)(0.5 * (2.0 * 1.3 / 128.0)); }
__device__ __forceinline__ float k_CB()   { return (float)(1.3 - 0.5 * (2.0 * 1.3 / 128.0)); }

// Per-axis trilinear setup, replicating the reference op-for-op.
__device__ __forceinline__ void axis_coords(float p, int& i0, int& i1, float& fr,
                                            int& ob0, int& ob1) {
  const float G = k_G(), VL = k_VL(), HALF = k_HALF(), CB = k_CB();
  const float EPSF = 1e-10f;
  float n0 = fminf(fmaxf(p - HALF, -G), G);
  float n1 = fminf(fmaxf(p + HALF, -G), G);
  float c0 = fminf(fmaxf((floorf(n0 / VL + EPSF) + 0.5f) * VL, -CB), CB);
  float c1 = fminf(fmaxf((floorf(n1 / VL + EPSF) + 0.5f) * VL, -CB), CB);
  int a0 = (int)floorf(c0 / VL + EPSF) + RESO / 2;
  int a1 = (int)floorf(c1 / VL + EPSF) + RESO / 2;
  ob0 = (a0 < 0) | (a0 > RESO - 1);
  ob1 = (a1 < 0) | (a1 > RESO - 1);
  i0 = min(max(a0, 0), RESO - 1);
  i1 = min(max(a1, 0), RESO - 1);
  fr = (p - c0) / VL;
}

__global__ __launch_bounds__(256) void nerf_march_kernel(
    const float* __restrict__ rays,   // (192,192,6)
    const float* __restrict__ dgrid,  // (128,128,128,1)
    const float* __restrict__ cgrid,  // (128,128,128,3)
    float* __restrict__ out) {        // (192,192,3)
  // One 16x16 pixel tile per block.
  const int bx = blockIdx.x % (IMG / TILE);
  const int by = blockIdx.x / (IMG / TILE);
  const int tx = threadIdx.x & 15;
  const int ty = threadIdx.x >> 4;
  const int px_x = bx * TILE + tx;
  const int px_y = by * TILE + ty;
  const int pix  = px_y * IMG + px_x;

  // ---- Stage the block's ray tile into LDS via the Tensor Data Mover ----
  // Tile: 16 rows x 96 f32 (row stride 1152 f32) -> 6 KB in LDS.
  __shared__ float lds_rays[TILE * ROWF];   // single static alloc -> LDS off 0

#if __has_builtin(__builtin_amdgcn_tensor_load_to_lds)
  if (threadIdx.x < 32) {                   // one wave issues the DMA
    const unsigned long long gaddr =
        (unsigned long long)(const void*)(rays +
            (size_t)(by * TILE) * IMG_ROWF + (size_t)bx * ROWF);
    // D# group 0: count=1 | lds_addr | global_addr[56:0] | type=2
    u32x4 g0;
    g0[0] = 1u;                                             // count=1 (valid)
    g0[1] = 0u;                                             // lds_addr = 0
    g0[2] = (unsigned int)(gaddr & 0xffffffffull);          // addr[31:0]
    g0[3] = (unsigned int)((gaddr >> 32) & 0x01ffffffull)   // addr[56:32]
            | (2u << 30);                                   // type=2 (image)
    // D# group 1: data_size=4B, tensor 1152x192, tile 96x16, stride 1152
    i32x8 g1;
    g1[0] = (2 << 16);                                      // data_size=4B
    g1[1] = (int)((IMG_ROWF & 0xffff) << 16);               // tensor_dim0 lo16
    g1[2] = (int)((IMG_ROWF >> 16) | (IMG << 16));          // dim0 hi16 | dim1 lo16
    g1[3] = (int)((IMG >> 16) | (ROWF << 16));              // dim1 hi16 | tile_dim0
    g1[4] = (int)(TILE);                                    // tile_dim1 | tile_dim2=0
    g1[5] = (int)(IMG_ROWF);                                // dim0_stride[31:0]
    g1[6] = 0;                                              // stride hi | dim1_stride lo
    g1[7] = 0;
    i32x4 gz4 = {0, 0, 0, 0};                               // groups 2/3: unused dims
    i32x8 gz8 = {0, 0, 0, 0, 0, 0, 0, 0};                   // extra group (unused)
    __builtin_amdgcn_tensor_load_to_lds(g0, g1, gz4, gz4, gz8, 0);
    __builtin_amdgcn_s_wait_tensorcnt(0);
  }
  // Optimizer fence: clang does not model the TDM builtin as writing LDS, so
  // without this the lds_rays reads fold to undef and the march loop is DCE'd.
  // Emits no instructions; just makes the compiler assume lds_rays was written.
  asm volatile("" : : "r"(&lds_rays[0]) : "memory");
  __syncthreads();
  const float* rp = lds_rays + ty * ROWF + tx * 6;
#else
  const float* rp = rays + (size_t)pix * 6;
#endif

  const float rox = rp[0], roy = rp[1], roz = rp[2];
  const float rdx = rp[3], rdy = rp[4], rdz = rp[5];
  const float dn  = sqrtf(rdx * rdx + rdy * rdy + rdz * rdz);

  const float T0 = 0.2f;
  const float DT = (float)(1.8 / 63.0);              // linspace step
  const float dist_reg = fmaxf(DT, 1e-5f) * dn;      // per-step dist * |r_d|
  const float VL = k_VL();

  float T = 1.0f;                                    // exclusive transmittance
  float aR = 0.0f, aG = 0.0f, aB = 0.0f;

#pragma unroll 2
  for (int s = 0; s < NSTEP; ++s) {
    const float t  = fmaf((float)s, DT, T0);
    const float px = fmaf(t, rdx, rox);
    const float py = fmaf(t, rdy, roy);
    const float pz = fmaf(t, rdz, roz);

    // gfx1250 prefetch of next step's base-corner cachelines: step s+1
    // addresses don't depend on step s's composite, so this hides L2 gather
    // latency behind the sequential cumprod dependency chain.
    if (s + 1 < NSTEP) {
      const float tn = t + DT;
      int qx = min(max((int)floorf(fmaf(tn, rdx, rox) / VL) + 64, 0), RESO - 1);
      int qy = min(max((int)floorf(fmaf(tn, rdy, roy) / VL) + 64, 0), RESO - 1);
      int qz = min(max((int)floorf(fmaf(tn, rdz, roz) / VL) + 64, 0), RESO - 1);
      int qv = (qx * RESO + qy) * RESO + qz;
      __builtin_prefetch(dgrid + qv, 0, 3);          // -> global_prefetch_b8
      __builtin_prefetch(cgrid + (size_t)3 * qv, 0, 3);
    }

    int ix0, ix1, iy0, iy1, iz0, iz1;
    int obx0, obx1, oby0, oby1, obz0, obz1;
    float fx, fy, fz;
    axis_coords(px, ix0, ix1, fx, obx0, obx1);
    axis_coords(py, iy0, iy1, fy, oby0, oby1);
    axis_coords(pz, iz0, iz1, fz, obz0, obz1);

    const float gx[2] = {1.0f - fx, fx};
    const float gy[2] = {1.0f - fy, fy};
    const float gz2[2] = {1.0f - fz, fz};
    const int   jx[2] = {ix0, ix1}, jy[2] = {iy0, iy1}, jz[2] = {iz0, iz1};
    const int   mx[2] = {obx0, obx1}, my[2] = {oby0, oby1}, mz[2] = {obz0, obz1};

    float dsum = 0.0f, rs = 0.0f, gs = 0.0f, bs = 0.0f;
    // Fully unrolled: all 32 gathers issue back-to-back (max MLP per wave).
#pragma unroll
    for (int u = 0; u < 8; ++u) {
      const int b0 = u & 1, b1 = (u >> 1) & 1, b2 = (u >> 2) & 1;
      float w = (gx[b0] * gy[b1]) * gz2[b2];
      if (mx[b0] | my[b1] | mz[b2]) w = 0.0f;        // reference oob zeroing
      const int vi = (jx[b0] * RESO + jy[b1]) * RESO + jz[b2];
      dsum = fmaf(w, dgrid[vi], dsum);
      const float* c = cgrid + (size_t)3 * vi;
      rs = fmaf(w, c[0], rs);
      gs = fmaf(w, c[1], gs);
      bs = fmaf(w, c[2], bs);
    }

    // sigmoid(rgb), softplus(density - 1) (numerically-stable forms)
    const float sr = 1.0f / (1.0f + expf(-rs));
    const float sg = 1.0f / (1.0f + expf(-gs));
    const float sb = 1.0f / (1.0f + expf(-bs));
    const float xa = dsum - 1.0f;
    const float sa = fmaxf(xa, 0.0f) + log1pf(expf(-fabsf(xa)));

    const float dist  = (s == NSTEP - 1) ? (1e10f * dn) : dist_reg;
    const float alpha = 1.0f - expf(-sa * dist);
    const float wgt   = alpha * T;
    aR = fmaf(wgt, sr, aR);
    aG = fmaf(wgt, sg, aG);
    aB = fmaf(wgt, sb, aB);
    T  = T * (1.0f - alpha + 1e-10f);                // cumprod with +1e-10
  }

  float* op = out + (size_t)pix * 3;
  op[0] = aR;
  op[1] = aG;
  op[2] = aB;
}

extern "C" void kernel_launch(void* const* d_in, const int* in_sizes, int n_in,
                              void* d_out, int out_size, void* d_ws, size_t ws_size,
                              hipStream_t stream) {
  (void)in_sizes; (void)n_in; (void)d_ws; (void)ws_size; (void)out_size;
  const float* rays  = (const float*)d_in[0];  // (1,192,192,6) f32
  const float* dens  = (const float*)d_in[1];  // (128,128,128,1) f32
  const float* rgbg  = (const float*)d_in[2];  // (128,128,128,3) f32
  float* out = (float*)d_out;                  // (1,192,192,3) f32

  dim3 block(256);                             // 8 wave32s per block
  dim3 grid((IMG / TILE) * (IMG / TILE));      // 144 blocks of 16x16 pixels
  hipLaunchKernelGGL(nerf_march_kernel, grid, block, 0, stream,
                     rays, dens, rgbg, out);
}